// Sampler_84507776516829
// MI455X (gfx1250) — compile-verified
//
#include <hip/hip_runtime.h>
#include <hip/hip_bf16.h>
#include <stdint.h>

// ---------------- problem constants (from reference setup_inputs) -----------
#define VOCAB   128000
#define ROWS    256
#define TPB     256          // 8 wave32s per block
#define TILE    1024         // floats per LDS tile (256 threads * 4 floats)
#define NTILES  (VOCAB / TILE)   // 125, exact
#define NBUF    4            // async pipeline buffers
#define DEPTH   3            // prefetch distance (s_wait_asynccnt 3 steady state)
#define NBINS   512
#define TOPN    20
#define LOG2E   1.4426950408889634f
#define LN2     0.6931471805599453f

typedef __attribute__((ext_vector_type(2))) float v2f;
typedef __attribute__((ext_vector_type(8))) float v8f;

#if defined(__has_builtin)
#  if __has_builtin(__builtin_amdgcn_wmma_f32_16x16x4_f32)
#    define HAVE_WMMA_F32X4 1
#  endif
#  if __has_builtin(__builtin_amdgcn_exp2f)
#    define FAST_EXP2(x) __builtin_amdgcn_exp2f(x)
#  endif
#  if __has_builtin(__builtin_amdgcn_logf)
#    define FAST_LOG2(x) __builtin_amdgcn_logf(x)
#  endif
#endif
#ifndef FAST_EXP2
#  define FAST_EXP2(x) exp2f(x)
#endif
#ifndef FAST_LOG2
#  define FAST_LOG2(x) log2f(x)
#endif

// ---------------- CDNA5 async global->LDS streaming (pipelined) -------------
// SADDR form: per-lane VGPR holds a loop-invariant 32-bit byte offset, the
// uniform tile base lives in SGPRs and advances on the SALU.  Tracked by
// ASYNCcnt; steady state keeps DEPTH tiles in flight per wave.  Each thread
// reads only the LDS slots it loaded itself, so ASYNCcnt alone synchronizes
// (no barriers in the streaming loops).
//
// The tail uses CLAMPED DUPLICATE PREFETCH: past the end we re-issue the last
// tile into its own buffer (byte-identical writes -> benign), so the loop has
// a single body and a single literal wait -- no epilogue body copies (which
// round 3 showed cause VGPR-array spills) and no per-iteration branch ladder.
// A full drain at the end keeps leftover duplicate writes from racing the
// next pass's reuse of the buffers (async LDS writes may complete OOO).
template <typename Body>
__device__ __forceinline__ void stream_row(const float* __restrict__ row,
                                           uint32_t lds_t, int t, Body&& body) {
    const uint32_t voff = (uint32_t)t * 16u;
    auto issue = [&](int tile) {
        asm volatile("global_load_async_to_lds_b128 %0, %1, %2"
                     :: "v"(lds_t + ((uint32_t)(tile & (NBUF - 1)) << 12)),
                        "v"(voff),
                        "s"(row + (size_t)tile * TILE)
                     : "memory");
    };
    issue(0); issue(1); issue(2);                       // prime the pipe
    for (int tile = 0; tile < NTILES; ++tile) {
        int pre = tile + DEPTH;
        pre = (pre < NTILES) ? pre : (NTILES - 1);      // clamped duplicate prefetch
        issue(pre);
        asm volatile("s_wait_asynccnt 3" ::: "memory"); // tile's load is complete
        body(tile);
    }
    asm volatile("s_wait_asynccnt 0" ::: "memory");     // drain before buffer reuse
}

// ---------------- helpers ----------------------------------------------------
__device__ __forceinline__ float rng_u01(uint32_t s) {
    s ^= s >> 16; s *= 0x7feb352du;
    s ^= s >> 15; s *= 0x846ca68bu;
    s ^= s >> 16;
    return (float)(s >> 8) * (1.0f / 16777216.0f) + 1.0e-20f;
}

#ifdef HAVE_WMMA_F32X4
// Wave-level f32 sum held in a 16x16 f32 WMMA accumulator (B == ones):
// every column of D is identical; lanes 0-15 hold rows 0-7 in c[0..7],
// lanes 16-31 hold rows 8-15.  total = colsum(lane) + colsum(lane^16).
__device__ __forceinline__ float wave_total_from_acc(v8f acc) {
    float p = acc[0] + acc[1] + acc[2] + acc[3] + acc[4] + acc[5] + acc[6] + acc[7];
    p += __shfl_xor(p, 16);
    return p;
}
#endif

__global__ __launch_bounds__(TPB)
void Sampler_84507776516829_kernel(const float* __restrict__ logits,
                                   const float* __restrict__ temperature,
                                   const float* __restrict__ top_p,
                                   const int*   __restrict__ top_k,
                                   int*   __restrict__ sampled_out,
                                   float* __restrict__ lp_out,
                                   int*   __restrict__ idx_out)
{
    __shared__ float    buf[NBUF][TILE];      // 16 KB async pipeline (depth 3)
    __shared__ uint32_t hcnt[NBINS];          // 2 KB
    __shared__ float    hmass[NBINS];         // 2 KB
    __shared__ float    s_red[16];
    __shared__ int      s_redi[8];
    __shared__ float    s_f0, s_f1;
    __shared__ int      s_i0, s_thr, s_win;
    __shared__ float    cand_v[TPB * TOPN];   // 20 KB
    __shared__ int      cand_i[TPB * TOPN];   // 20 KB

    const int b    = blockIdx.x;
    const int t    = threadIdx.x;
    const int lane = t & 31;
    const int wave = t >> 5;
    const float* row = logits + (size_t)b * VOCAB;

    // raw LDS byte address of this thread's slot in buffer 0 (buffer k: +k<<12)
    const uint32_t lds_t = (uint32_t)(uintptr_t)(&buf[0][0]) + (uint32_t)t * 16u;

    // clear histogram (made visible by the syncs at end of pass 1)
    for (int i = t; i < NBINS; i += TPB) { hcnt[i] = 0u; hmass[i] = 0.0f; }
    if (t == 0) s_thr = NBINS;

    // ============ Pass 1: row max + argmax (async pipelined stream) =========
    float m  = -__builtin_inff();
    int   am = 0;
    stream_row(row, lds_t, t, [&](int tile) {
        const float4 x = *reinterpret_cast<const float4*>(&buf[tile & (NBUF - 1)][t * 4]);
        const int base = tile * TILE + t * 4;
        if (x.x > m) { m = x.x; am = base;     }
        if (x.y > m) { m = x.y; am = base + 1; }
        if (x.z > m) { m = x.z; am = base + 2; }
        if (x.w > m) { m = x.w; am = base + 3; }
    });
    for (int d = 16; d > 0; d >>= 1) {           // wave32 argmax
        float ov = __shfl_xor(m, d);
        int   oi = __shfl_xor(am, d);
        if (ov > m || (ov == m && oi < am)) { m = ov; am = oi; }
    }
    if (lane == 0) { s_red[wave] = m; s_redi[wave] = am; }
    __syncthreads();
    if (t == 0) {
        float bm = s_red[0]; int bi = s_redi[0];
        for (int w = 1; w < TPB / 32; ++w)
            if (s_red[w] > bm || (s_red[w] == bm && s_redi[w] < bi)) { bm = s_red[w]; bi = s_redi[w]; }
        s_f0 = bm; s_i0 = bi;
    }
    __syncthreads();
    m = s_f0;
    const int argmax_idx = s_i0;
    __syncthreads();

    float temp = temperature[b];
    const bool greedy = (temp < 1e-5f);
    if (greedy) temp = 1.0f;
    const float invT = 1.0f / temp;

    // ====== Pass 2: partition sums (WMMA reduction) + log-prob histogram ====
#ifdef HAVE_WMMA_F32X4
    v8f acc_o = {};                 // sum exp(x - m)              (orig)
    v8f acc_s = {};                 // sum exp((x - m) * invT)     (scaled)
    const v2f ones = {1.0f, 1.0f};
#else
    float acc_o = 0.0f, acc_s = 0.0f;
#endif
    stream_row(row, lds_t, t, [&](int tile) {
        const float4 x = *reinterpret_cast<const float4*>(&buf[tile & (NBUF - 1)][t * 4]);
        const float d0 = x.x - m, d1 = x.y - m, d2 = x.z - m, d3 = x.w - m;
        // raw v_exp_f32: tails underflow to 0, which is exactly what we want
        const float eo0 = FAST_EXP2(d0 * LOG2E), eo1 = FAST_EXP2(d1 * LOG2E);
        const float eo2 = FAST_EXP2(d2 * LOG2E), eo3 = FAST_EXP2(d3 * LOG2E);
        const float u0 = -d0 * invT * LOG2E, u1 = -d1 * invT * LOG2E;
        const float u2 = -d2 * invT * LOG2E, u3 = -d3 * invT * LOG2E;
        const float es0 = FAST_EXP2(-u0), es1 = FAST_EXP2(-u1);
        const float es2 = FAST_EXP2(-u2), es3 = FAST_EXP2(-u3);
        // histogram of log2-prob deficit; bins past the scan horizon can never
        // change the threshold, so skip them (kills the same-address ds_add
        // hot-spot when invT is huge)
        const float bf0 = u0 * 8.0f, bf1 = u1 * 8.0f, bf2 = u2 * 8.0f, bf3 = u3 * 8.0f;
        if (bf0 < (float)(NBINS - 1)) { int i0 = (int)bf0; atomicAdd(&hcnt[i0], 1u); atomicAdd(&hmass[i0], es0); }
        if (bf1 < (float)(NBINS - 1)) { int i1 = (int)bf1; atomicAdd(&hcnt[i1], 1u); atomicAdd(&hmass[i1], es1); }
        if (bf2 < (float)(NBINS - 1)) { int i2 = (int)bf2; atomicAdd(&hcnt[i2], 1u); atomicAdd(&hmass[i2], es2); }
        if (bf3 < (float)(NBINS - 1)) { int i3 = (int)bf3; atomicAdd(&hcnt[i3], 1u); atomicAdd(&hmass[i3], es3); }
#ifdef HAVE_WMMA_F32X4
        v2f ao = { eo0 + eo1, eo2 + eo3 };
        v2f as = { es0 + es1, es2 + es3 };
        // D = A x ones + C : f32 row-sum reduction on the matrix pipe
        acc_o = __builtin_amdgcn_wmma_f32_16x16x4_f32(false, ao, false, ones,
                                                      (short)0, acc_o, false, false);
        acc_s = __builtin_amdgcn_wmma_f32_16x16x4_f32(false, as, false, ones,
                                                      (short)0, acc_s, false, false);
#else
        acc_o += eo0 + eo1 + eo2 + eo3;
        acc_s += es0 + es1 + es2 + es3;
#endif
    });
#ifdef HAVE_WMMA_F32X4
    float zo = wave_total_from_acc(acc_o);
    float zs = wave_total_from_acc(acc_s);
#else
    float zo = acc_o, zs = acc_s;
    for (int d = 16; d > 0; d >>= 1) { zo += __shfl_xor(zo, d); zs += __shfl_xor(zs, d); }
#endif
    if (lane == 0) { s_red[wave] = zo; s_red[8 + wave] = zs; }
    __syncthreads();
    if (t == 0) {
        float a = 0.0f, c = 0.0f;
        for (int w = 0; w < 8; ++w) { a += s_red[w]; c += s_red[8 + w]; }
        s_f0 = a; s_f1 = c;
    }
    __syncthreads();
    const float Zo = s_f0;
    const float Zs = s_f1;

    // ---- parallel inclusive scan of histogram (Hillis-Steele, in place) ----
    for (int s = 1; s < NBINS; s <<= 1) {
        const int i0 = t, i1 = t + TPB;
        const uint32_t c0 = hcnt[i0] + ((i0 >= s) ? hcnt[i0 - s] : 0u);
        const float    m0 = hmass[i0] + ((i0 >= s) ? hmass[i0 - s] : 0.0f);
        const uint32_t c1 = hcnt[i1] + ((i1 >= s) ? hcnt[i1 - s] : 0u);
        const float    m1 = hmass[i1] + ((i1 >= s) ? hmass[i1 - s] : 0.0f);
        __syncthreads();
        hcnt[i0] = c0; hmass[i0] = m0;
        hcnt[i1] = c1; hmass[i1] = m1;
        __syncthreads();
    }
    // threshold = first bin whose EXCLUSIVE prefix hits k or top_p mass
    {
        const float ptarget = top_p[b] * Zs;
        const uint32_t k = (uint32_t)max(top_k[b], 1);
        for (int i = t; i < NBINS; i += TPB) {
            if (i >= 1 && (hcnt[i - 1] >= k || hmass[i - 1] >= ptarget))
                atomicMin(&s_thr, i);
        }
    }
    __syncthreads();
    const float thrf = (float)s_thr;   // >= 1: bin 0 (the max) always kept

    // ====== Pass 3: Gumbel-max over kept set + per-thread top-20 ============
    float best = -__builtin_inff();
    int   bestidx = 0;
    float lv[TOPN]; int li[TOPN];
#pragma unroll
    for (int j = 0; j < TOPN; ++j) { lv[j] = -__builtin_inff(); li[j] = 0; }

    auto proc = [&](float xv, int vi) {
        const float ue = (m - xv) * invT * LOG2E;       // >= 0, log2 prob deficit
        if (ue * 8.0f < thrf) {                          // inside top-k/top-p set
            const float u = rng_u01(((uint32_t)b * 1315423911u) ^ ((uint32_t)vi * 2654435761u));
            const float g = -LN2 * FAST_LOG2(-LN2 * FAST_LOG2(u));
            const float sc = xv * invT + g;
            if (sc > best) { best = sc; bestidx = vi; }
        }
        if (xv > lv[TOPN - 1]) {                         // top-20 insertion (unrolled)
            float cv = xv; int ci = vi;
#pragma unroll
            for (int j = 0; j < TOPN; ++j) {
                if (cv > lv[j]) {
                    float tv = lv[j]; int ti = li[j];
                    lv[j] = cv; li[j] = ci; cv = tv; ci = ti;
                }
            }
        }
    };
    stream_row(row, lds_t, t, [&](int tile) {
        const float4 x = *reinterpret_cast<const float4*>(&buf[tile & (NBUF - 1)][t * 4]);
        const int base = tile * TILE + t * 4;
        proc(x.x, base); proc(x.y, base + 1); proc(x.z, base + 2); proc(x.w, base + 3);
    });

    // ---- block argmax of gumbel scores -> sampled token --------------------
    for (int d = 16; d > 0; d >>= 1) {
        float ov = __shfl_xor(best, d);
        int   oi = __shfl_xor(bestidx, d);
        if (ov > best || (ov == best && oi < bestidx)) { best = ov; bestidx = oi; }
    }
    if (lane == 0) { s_red[wave] = best; s_redi[wave] = bestidx; }
    __syncthreads();
    if (t == 0) {
        float bv = s_red[0]; int bi = s_redi[0];
        for (int w = 1; w < 8; ++w)
            if (s_red[w] > bv || (s_red[w] == bv && s_redi[w] < bi)) { bv = s_red[w]; bi = s_redi[w]; }
        sampled_out[b] = greedy ? argmax_idx : bi;
    }
    __syncthreads();

    // ---- merge per-thread top-20 into block top-20 -------------------------
#pragma unroll
    for (int j = 0; j < TOPN; ++j) {
        cand_v[t * TOPN + j] = lv[j];
        cand_i[t * TOPN + j] = li[j];
    }
    __syncthreads();

    const float lnZo = LN2 * FAST_LOG2(Zo);
    int ptr = 0;
    for (int r = 0; r < TOPN; ++r) {
        const float cv = cand_v[t * TOPN + ptr];   // each thread's current max
        const int   ci = cand_i[t * TOPN + ptr];
        float v = cv; int who = t;
        for (int d = 16; d > 0; d >>= 1) {
            float ov = __shfl_xor(v, d);
            int   ow = __shfl_xor(who, d);
            if (ov > v || (ov == v && ow < who)) { v = ov; who = ow; }
        }
        if (lane == 0) { s_red[wave] = v; s_redi[wave] = who; }
        __syncthreads();
        if (t == 0) {
            float bv = s_red[0]; int bw = s_redi[0];
            for (int w = 1; w < 8; ++w)
                if (s_red[w] > bv || (s_red[w] == bv && s_redi[w] < bw)) { bv = s_red[w]; bw = s_redi[w]; }
            s_win = bw;
        }
        __syncthreads();
        if (t == s_win) {
            lp_out[b * TOPN + r]  = cv - m - lnZo;   // log_softmax of original logits
            idx_out[b * TOPN + r] = ci;
            ++ptr;                                    // pop my max
        }
    }
}

extern "C" void kernel_launch(void* const* d_in, const int* in_sizes, int n_in,
                              void* d_out, int out_size, void* d_ws, size_t ws_size,
                              hipStream_t stream) {
    const float* logits = (const float*)d_in[0];
    const float* temp   = (const float*)d_in[1];
    const float* topp   = (const float*)d_in[2];
    const int*   topk   = (const int*)d_in[3];

    const int rows = (n_in > 1) ? in_sizes[1] : ROWS;   // 256

    float* out_f   = (float*)d_out;
    int*   sampled = (int*)out_f;                        // [rows] int32 (bit-cast slots)
    float* lp      = out_f + rows;                       // [rows*20] f32
    int*   idx     = (int*)(out_f + rows + rows * TOPN); // [rows*20] int32

    Sampler_84507776516829_kernel<<<dim3(rows), dim3(TPB), 0, stream>>>(
        logits, temp, topp, topk, sampled, lp, idx);
}